// cofeature_fast_60361470378211
// MI455X (gfx1250) — compile-verified
//
#include <hip/hip_runtime.h>

typedef __attribute__((ext_vector_type(2))) float v2f;
typedef __attribute__((ext_vector_type(8))) float v8f;

constexpr int NB  = 8;     // batch
constexpr int NC  = 512;   // channels
constexpr int NS  = 196;   // 14*14 spatial
constexpr int NHS = 1024;  // hidden
constexpr int NK  = 144;   // 12*12 interior centers
constexpr int NO  = 9;     // offsets

// ---------------------------------------------------------------------------
// Phase A (WMMA): P[b,s,h] = sum_c x[b,c,s]*w[h,c] + bias[h], for (w1,b1) and
// (w2,b2) simultaneously (shared A fragments from x).
// GEMM per b: M = s (196 -> 13 tiles of 16), N = h (1024), K = c (512).
//   A[m=s,k=c] = x[b,c,s]  -> lanes index s: coalesced b32 loads,
//                             branchless 0-padding via clamp + cndmask
//   B[k=c,n=h] = w[h,c]    -> per lane (w[h,k], w[h,k+1]): one float2 load
// Each wave: 16 s x 32 h for BOTH matrices (4 v8f accumulators).
// grid = (13 s-tiles, NB, 2 h-halves), block = 512 (16 waves, 512 h / block)
// ---------------------------------------------------------------------------
__global__ __launch_bounds__(512) void proj_wmma_kernel(
    const float* __restrict__ x,
    const float* __restrict__ w1, const float* __restrict__ b1,
    const float* __restrict__ w2, const float* __restrict__ b2,
    float* __restrict__ P1, float* __restrict__ P2)
{
    const int tid  = threadIdx.x;
    const int lane = tid & 31;
    const int wave = tid >> 5;                 // 0..15
    const int half = lane >> 4;                // K-half selector
    const int l15  = lane & 15;

    const int s_base = blockIdx.x * 16;        // 0..192 (13 tiles, padded)
    const int b      = blockIdx.y;
    const int h_base = blockIdx.z * 512 + wave * 32;

    const int  scol   = s_base + l15;          // A: M = lane&15
    const bool sok    = (scol < NS);
    const int  sclamp = sok ? scol : (NS - 1); // always-valid address
    const float smask = sok ? 1.0f : 0.0f;

    // per-lane A base: x[b, 2*half, sclamp]; step over k0 via offset
    const float* xa = x + (size_t)b * NC * NS + (size_t)(2 * half) * NS + sclamp;

    // per-lane weight row pointers (B operand: col n = l15)
    const float* w1r0 = w1 + (size_t)(h_base +  0 + l15) * NC + 2 * half;
    const float* w1r1 = w1 + (size_t)(h_base + 16 + l15) * NC + 2 * half;
    const float* w2r0 = w2 + (size_t)(h_base +  0 + l15) * NC + 2 * half;
    const float* w2r1 = w2 + (size_t)(h_base + 16 + l15) * NC + 2 * half;

    v8f acc10 = {}, acc11 = {}, acc20 = {}, acc21 = {};

    for (int k0 = 0; k0 < NC; k0 += 4) {
        v2f a;                                 // A 16x4 fragment (from x)
        a.x = xa[(size_t)k0 * NS]      * smask;   // unconditional load + mul
        a.y = xa[(size_t)k0 * NS + NS] * smask;

        // B 4x16 fragments: consecutive K pair -> single float2 load each
        const float2 f10 = *(const float2*)(w1r0 + k0);
        const float2 f11 = *(const float2*)(w1r1 + k0);
        const float2 f20 = *(const float2*)(w2r0 + k0);
        const float2 f21 = *(const float2*)(w2r1 + k0);
        v2f b10; b10.x = f10.x; b10.y = f10.y;
        v2f b11; b11.x = f11.x; b11.y = f11.y;
        v2f b20; b20.x = f20.x; b20.y = f20.y;
        v2f b21; b21.x = f21.x; b21.y = f21.y;

        acc10 = __builtin_amdgcn_wmma_f32_16x16x4_f32(false, a, false, b10, (short)0, acc10, false, false);
        acc11 = __builtin_amdgcn_wmma_f32_16x16x4_f32(false, a, false, b11, (short)0, acc11, false, false);
        acc20 = __builtin_amdgcn_wmma_f32_16x16x4_f32(false, a, false, b20, (short)0, acc20, false, false);
        acc21 = __builtin_amdgcn_wmma_f32_16x16x4_f32(false, a, false, b21, (short)0, acc21, false, false);
    }

    // bias per output column (h), identical for all 8 rows held in a VGPR
    const float bv10 = b1[h_base +  0 + l15];
    const float bv11 = b1[h_base + 16 + l15];
    const float bv20 = b2[h_base +  0 + l15];
    const float bv21 = b2[h_base + 16 + l15];

    // D layout: VGPR r, lanes 0-15 -> row r, lanes 16-31 -> row r+8
#pragma unroll
    for (int r = 0; r < 8; ++r) {
        const int srow = s_base + r + 8 * half;
        if (srow < NS) {
            const size_t o0 = ((size_t)b * NS + srow) * NHS + h_base + l15;
            P1[o0]      = acc10[r] + bv10;
            P1[o0 + 16] = acc11[r] + bv11;
            P2[o0]      = acc20[r] + bv20;
            P2[o0 + 16] = acc21[r] + bv21;
        }
    }
}

// ---------------------------------------------------------------------------
// Phase B: cofe[b,o,i,j] = sum_k P1[b,cs(k),i] * P2[b,cs(k)+delta(o),j]
// then row-normalize over j.  One workgroup = one (b,o) x 16-row strip.
// 16 waves; each wave: 16(M) x 64(N) tile via 4x v8f accumulators,
// K=144 in 36 steps of V_WMMA_F32_16X16X4_F32.
// grid = NB*NO*64, block = 512 (16 wave32)
// ---------------------------------------------------------------------------
__global__ __launch_bounds__(512) void cofe_kernel(
    const float* __restrict__ P1, const float* __restrict__ P2,
    float* __restrict__ out)
{
    __shared__ float rowsum[16];
    __shared__ int   csLDS[NK];

    const int tid  = threadIdx.x;
    const int lane = tid & 31;
    const int wave = tid >> 5;           // 0..15
    const int bid  = blockIdx.x;
    const int strip = bid & 63;          // 64 strips of 16 rows
    const int bo    = bid >> 6;          // 0..71
    const int b = bo / NO;
    const int o = bo % NO;
    const int i_base = strip * 16;
    const int j_base = wave * 64;

    if (tid < NK) {
        int ky = tid / 12, kx = tid % 12;
        csLDS[tid] = (ky + 1) * 14 + (kx + 1);   // interior spatial index
    }
    if (tid < 16) rowsum[tid] = 0.0f;
    __syncthreads();

    const int delta = (o / 3 - 1) * 14 + (o % 3 - 1);

    const float* Ap = P1 + (size_t)b * NS * NHS;
    const float* Bp = P2 + (size_t)b * NS * NHS;

    v8f acc0 = {}, acc1 = {}, acc2 = {}, acc3 = {};

    const int half = lane >> 4;          // 0: K=k0,k0+1   1: K=k0+2,k0+3
    const int l15  = lane & 15;
    const int icol = i_base + l15;       // A: M = lane&15

    for (int k0 = 0; k0 < NK; k0 += 4) {
        const int ka  = k0 + 2 * half;
        const int sa0 = csLDS[ka];
        const int sa1 = csLDS[ka + 1];

        v2f a;                                            // A 16x4 fragment
        a.x = Ap[(size_t)sa0 * NHS + icol];
        a.y = Ap[(size_t)sa1 * NHS + icol];

        const float* Br0 = Bp + (size_t)(sa0 + delta) * NHS + j_base + l15;
        const float* Br1 = Bp + (size_t)(sa1 + delta) * NHS + j_base + l15;
        v2f b0;  b0.x  = Br0[0];   b0.y  = Br1[0];        // B 4x16 fragments
        v2f b1v; b1v.x = Br0[16];  b1v.y = Br1[16];
        v2f b2v; b2v.x = Br0[32];  b2v.y = Br1[32];
        v2f b3v; b3v.x = Br0[48];  b3v.y = Br1[48];

        acc0 = __builtin_amdgcn_wmma_f32_16x16x4_f32(false, a, false, b0,  (short)0, acc0, false, false);
        acc1 = __builtin_amdgcn_wmma_f32_16x16x4_f32(false, a, false, b1v, (short)0, acc1, false, false);
        acc2 = __builtin_amdgcn_wmma_f32_16x16x4_f32(false, a, false, b2v, (short)0, acc2, false, false);
        acc3 = __builtin_amdgcn_wmma_f32_16x16x4_f32(false, a, false, b3v, (short)0, acc3, false, false);
    }

    // --- fused row-norm: sum of squares per output row across the strip ---
    // C/D layout: VGPR r, lanes 0-15 -> row r, lanes 16-31 -> row r+8.
    float ssum[8];
#pragma unroll
    for (int r = 0; r < 8; ++r) {
        float t = acc0[r] * acc0[r] + acc1[r] * acc1[r] +
                  acc2[r] * acc2[r] + acc3[r] * acc3[r];
        t += __shfl_xor(t, 1, 32);
        t += __shfl_xor(t, 2, 32);
        t += __shfl_xor(t, 4, 32);
        t += __shfl_xor(t, 8, 32);       // full sum within each 16-lane half
        ssum[r] = t;
    }
    if (l15 == 0) {
#pragma unroll
        for (int r = 0; r < 8; ++r)
            atomicAdd(&rowsum[r + 8 * half], ssum[r]);   // ds_add_f32
    }
    __syncthreads();

    float inv[8];
#pragma unroll
    for (int r = 0; r < 8; ++r) {
        float n = sqrtf(rowsum[r + 8 * half]);
        inv[r] = 1.0f / fmaxf(n, 1e-12f);
    }

    // out[b][o][i][j]
    float* op = out + (((size_t)b * NO + o) * NHS + i_base) * (size_t)NHS;
#pragma unroll
    for (int r = 0; r < 8; ++r) {
        const int m = r + 8 * half;
        const size_t ro = (size_t)m * NHS + j_base + l15;
        op[ro]      = acc0[r] * inv[r];
        op[ro + 16] = acc1[r] * inv[r];
        op[ro + 32] = acc2[r] * inv[r];
        op[ro + 48] = acc3[r] * inv[r];
    }
}

// ---------------------------------------------------------------------------
extern "C" void kernel_launch(void* const* d_in, const int* in_sizes, int n_in,
                              void* d_out, int out_size, void* d_ws, size_t ws_size,
                              hipStream_t stream) {
    (void)in_sizes; (void)n_in; (void)out_size; (void)ws_size;
    const float* x  = (const float*)d_in[0];
    const float* w1 = (const float*)d_in[1];
    const float* b1 = (const float*)d_in[2];
    const float* w2 = (const float*)d_in[3];
    const float* b2 = (const float*)d_in[4];
    float* out = (float*)d_out;

    float* P1 = (float*)d_ws;                       // 8*196*1024 f32 = 6.42 MB
    float* P2 = P1 + (size_t)NB * NS * NHS;         // another 6.42 MB

    dim3 gA(13, NB, 2);                             // s-tiles, batch, h-halves
    proj_wmma_kernel<<<gA, 512, 0, stream>>>(x, w1, b1, w2, b2, P1, P2);

    cofe_kernel<<<dim3(NB * NO * (NHS / 16)), 512, 0, stream>>>(P1, P2, out);
}